// iot_uni_19902878449982
// MI455X (gfx1250) — compile-verified
//
#include <hip/hip_runtime.h>
#include <math.h>

#define NROWS 8192
#define KDIM  512
#define LOG2N 13            // NROWS = 1<<13
#define EPS   0.05f
#define LAMBDA_P 0.2f

typedef __attribute__((ext_vector_type(2))) float v2f;
typedef __attribute__((ext_vector_type(8))) float v8f;

// ---------------- row normalize: zn = z / max(||z||, 1e-8) ----------------
__global__ __launch_bounds__(256) void normalize_kernel(const float* __restrict__ z,
                                                        float* __restrict__ zn) {
    const int row = blockIdx.x;
    const float* src = z + (size_t)row * KDIM;
    float s = 0.f;
    for (int j = threadIdx.x; j < KDIM; j += 256) { float v = src[j]; s += v * v; }
    __shared__ float red[256];
    red[threadIdx.x] = s; __syncthreads();
    for (int off = 128; off; off >>= 1) {
        if (threadIdx.x < off) red[threadIdx.x] += red[threadIdx.x + off];
        __syncthreads();
    }
    const float inv = 1.f / fmaxf(sqrtf(red[0]), 1e-8f);
    float* dst = zn + (size_t)row * KDIM;
    for (int j = threadIdx.x; j < KDIM; j += 256) dst[j] = src[j] * inv;
}

// ---------------- cos = zn @ zn^T via V_WMMA_F32_16X16X4_F32 ----------------
// Register-blocked: each wave computes a 16x64 strip (4 C tiles), reusing one
// A fragment against 4 B fragments -> 4 WMMAs per 5 b64 loads.
// block: 256 threads = 8 waves = 8 strips; grid: (8192/64/8 = 16, 512)
__global__ __launch_bounds__(256) void gemm_cos_kernel(const float* __restrict__ zn,
                                                       float* __restrict__ P) {
    const int lane = threadIdx.x & 31;
    const int wave = threadIdx.x >> 5;
    const int rowBase = blockIdx.y * 16;
    const int colBase = (blockIdx.x * 8 + wave) * 64;
    const int m  = lane & 15;           // A row / B col within tile
    const int kh = (lane >> 4) << 1;    // lanes 16-31 hold K+2,K+3

    const float* aRow = zn + (size_t)(rowBase + m) * KDIM + kh;
    const float* bRow = zn + (size_t)(colBase + m) * KDIM + kh;  // tile 0; tiles 1..3 at +16*KDIM

    v8f acc0 = {}, acc1 = {}, acc2 = {}, acc3 = {};
#pragma unroll 4
    for (int kk = 0; kk < KDIM; kk += 4) {
        if ((kk & 63) == 0) {
            __builtin_prefetch(aRow + kk + 64, 0, 1);            // global_prefetch_b8
            __builtin_prefetch(bRow + kk + 64, 0, 1);
        }
        v2f a  = *(const v2f*)(aRow + kk);
        v2f b0 = *(const v2f*)(bRow + kk);
        v2f b1 = *(const v2f*)(bRow + 16 * KDIM + kk);
        v2f b2 = *(const v2f*)(bRow + 32 * KDIM + kk);
        v2f b3 = *(const v2f*)(bRow + 48 * KDIM + kk);
        // (neg_a, A, neg_b, B, c_mod, C, reuse_a, reuse_b)
        acc0 = __builtin_amdgcn_wmma_f32_16x16x4_f32(false, a, false, b0, (short)0, acc0, false, false);
        acc1 = __builtin_amdgcn_wmma_f32_16x16x4_f32(false, a, false, b1, (short)0, acc1, false, false);
        acc2 = __builtin_amdgcn_wmma_f32_16x16x4_f32(false, a, false, b2, (short)0, acc2, false, false);
        acc3 = __builtin_amdgcn_wmma_f32_16x16x4_f32(false, a, false, b3, (short)0, acc3, false, false);
    }
    // C layout: VGPR v -> row M = v + 8*(lane>=16); col N = lane & 15
    const int rOut = rowBase + ((lane >> 4) << 3);
    const int cOut = colBase + (lane & 15);
#pragma unroll
    for (int v = 0; v < 8; ++v) {
        float* dst = P + (size_t)(rOut + v) * NROWS + cOut;
        dst[0]  = acc0[v];
        dst[16] = acc1[v];
        dst[32] = acc2[v];
        dst[48] = acc3[v];
    }
}

// ---------------- global min of cos (two-stage, deterministic) ----------------
__global__ __launch_bounds__(256) void min_partial_kernel(const float* __restrict__ P,
                                                          float* __restrict__ part) {
    const float4* __restrict__ P4 = (const float4*)P;
    const size_t total4 = (size_t)NROWS * NROWS / 4;
    const size_t stride = (size_t)gridDim.x * 256;
    float m = 3.4e38f;
    for (size_t i = (size_t)blockIdx.x * 256 + threadIdx.x; i < total4; i += stride) {
        float4 v = P4[i];
        m = fminf(m, fminf(fminf(v.x, v.y), fminf(v.z, v.w)));
    }
    __shared__ float red[256];
    red[threadIdx.x] = m; __syncthreads();
    for (int off = 128; off; off >>= 1) {
        if (threadIdx.x < off) red[threadIdx.x] = fminf(red[threadIdx.x], red[threadIdx.x + off]);
        __syncthreads();
    }
    if (threadIdx.x == 0) part[blockIdx.x] = red[0];
}

__global__ __launch_bounds__(256) void min_final_kernel(const float* __restrict__ part,
                                                        float* __restrict__ cmax, int nparts) {
    float m = 3.4e38f;
    for (int i = threadIdx.x; i < nparts; i += 256) m = fminf(m, part[i]);
    __shared__ float red[256];
    red[threadIdx.x] = m; __syncthreads();
    for (int off = 128; off; off >>= 1) {
        if (threadIdx.x < off) red[threadIdx.x] = fminf(red[threadIdx.x], red[threadIdx.x + off]);
        __syncthreads();
    }
    if (threadIdx.x == 0) cmax[0] = 1.0f - red[0];   // Cmax = max(1 - cos)
}

// ---------------- Probs = exp(-Cost/eps), diag = exp(-Cmax/eps) ----------------
__global__ __launch_bounds__(256) void exp_kernel(float* __restrict__ P,
                                                  const float* __restrict__ cmaxPtr) {
    const float diagVal = expf(-cmaxPtr[0] / EPS);
    const size_t total  = (size_t)NROWS * NROWS;
    const size_t stride = (size_t)gridDim.x * 256;
    for (size_t idx = (size_t)blockIdx.x * 256 + threadIdx.x; idx < total; idx += stride) {
        const int i = (int)(idx >> LOG2N);
        const int j = (int)(idx & (NROWS - 1));
        const float cosv = P[idx];
        P[idx] = (i == j) ? diagVal : expf(-(1.0f - cosv) / EPS);
    }
}

// ---------------- axis reductions ----------------
// out[row] = sum_j w[j] * P[row][j]   (w==nullptr -> 1)
__global__ __launch_bounds__(256) void rowsum_kernel(const float* __restrict__ P,
                                                     const float* __restrict__ w,
                                                     float* __restrict__ out) {
    const int row = blockIdx.x;
    const float4* __restrict__ src4 = (const float4*)(P + (size_t)row * NROWS);
    const float4* __restrict__ w4 = (const float4*)w;
    float s = 0.f;
    if (w) {
        for (int j = threadIdx.x; j < NROWS / 4; j += 256) {
            float4 v = src4[j]; float4 wv = w4[j];
            s += wv.x * v.x + wv.y * v.y + wv.z * v.z + wv.w * v.w;
        }
    } else {
        for (int j = threadIdx.x; j < NROWS / 4; j += 256) {
            float4 v = src4[j];
            s += v.x + v.y + v.z + v.w;
        }
    }
    __shared__ float red[256];
    red[threadIdx.x] = s; __syncthreads();
    for (int off = 128; off; off >>= 1) {
        if (threadIdx.x < off) red[threadIdx.x] += red[threadIdx.x + off];
        __syncthreads();
    }
    if (threadIdx.x == 0) out[row] = red[0];
}

// partial[by][4 cols per thread] = sum over 128-row chunk of w[i]*P[i][j]; grid (8, 64)
__global__ __launch_bounds__(256) void colsum_partial_kernel(const float* __restrict__ P,
                                                             const float* __restrict__ w,
                                                             float* __restrict__ part) {
    const int j4 = blockIdx.x * 256 + threadIdx.x;   // 0..2047 (each = 4 columns)
    const int i0 = blockIdx.y * 128;
    float4 s = make_float4(0.f, 0.f, 0.f, 0.f);
    if (w) {
        for (int i = i0; i < i0 + 128; ++i) {
            float4 v = ((const float4*)(P + (size_t)i * NROWS))[j4];
            const float wi = w[i];
            s.x += wi * v.x; s.y += wi * v.y; s.z += wi * v.z; s.w += wi * v.w;
        }
    } else {
        for (int i = i0; i < i0 + 128; ++i) {
            float4 v = ((const float4*)(P + (size_t)i * NROWS))[j4];
            s.x += v.x; s.y += v.y; s.z += v.z; s.w += v.w;
        }
    }
    ((float4*)(part + (size_t)blockIdx.y * NROWS))[j4] = s;
}

__global__ __launch_bounds__(256) void colsum_finish_kernel(const float* __restrict__ part,
                                                            float* __restrict__ out) {
    const int j4 = blockIdx.x * 256 + threadIdx.x;   // 0..2047
    float4 s = make_float4(0.f, 0.f, 0.f, 0.f);
    for (int b = 0; b < 64; ++b) {
        float4 v = ((const float4*)(part + (size_t)b * NROWS))[j4];
        s.x += v.x; s.y += v.y; s.z += v.z; s.w += v.w;
    }
    ((float4*)out)[j4] = s;
}

__global__ __launch_bounds__(256) void recip_kernel(const float* __restrict__ in,
                                                    float* __restrict__ out) {
    const int i = blockIdx.x * 256 + threadIdx.x;
    out[i] = 1.0f / in[i];
}

// P[a][b] *= u[a] * v[b]  (float4 along rows)
__global__ __launch_bounds__(256) void scale_kernel(float* __restrict__ P,
                                                    const float* __restrict__ u,
                                                    const float* __restrict__ v) {
    float4* __restrict__ P4 = (float4*)P;
    const float4* __restrict__ v4 = (const float4*)v;
    const size_t total4 = (size_t)NROWS * NROWS / 4;
    const size_t stride = (size_t)gridDim.x * 256;
    for (size_t idx = (size_t)blockIdx.x * 256 + threadIdx.x; idx < total4; idx += stride) {
        const int a  = (int)(idx >> (LOG2N - 2));          // row
        const int b4 = (int)(idx & (NROWS / 4 - 1));       // 4-col group
        const float ua = u[a];
        float4 vv = v4[b4];
        float4 p = P4[idx];
        p.x *= ua * vv.x; p.y *= ua * vv.y; p.z *= ua * vv.z; p.w *= ua * vv.w;
        P4[idx] = p;
    }
}

// ---------------- final losses (two-stage, double accumulators) ----------------
__global__ __launch_bounds__(256) void loss_partial_kernel(const float* __restrict__ P,
                                                           double* __restrict__ part) {
    const int HALF = NROWS / 2;
    const size_t total  = (size_t)NROWS * NROWS;
    const size_t stride = (size_t)gridDim.x * 256;
    double sP = 0.0, sLogOff = 0.0, sLogDiag = 0.0;
    for (size_t idx = (size_t)blockIdx.x * 256 + threadIdx.x; idx < total; idx += stride) {
        const int i = (int)(idx >> LOG2N);
        const int j = (int)(idx & (NROWS - 1));
        if (i == j) continue;
        const float p = P[idx];
        const float lp = logf(p);
        sP += (double)p;
        sLogOff += (double)lp;
        if (j == i + HALF || i == j + HALF) sLogDiag += (double)lp;
    }
    __shared__ double r0[256], r1[256], r2[256];
    r0[threadIdx.x] = sP; r1[threadIdx.x] = sLogOff; r2[threadIdx.x] = sLogDiag;
    __syncthreads();
    for (int off = 128; off; off >>= 1) {
        if (threadIdx.x < off) {
            r0[threadIdx.x] += r0[threadIdx.x + off];
            r1[threadIdx.x] += r1[threadIdx.x + off];
            r2[threadIdx.x] += r2[threadIdx.x + off];
        }
        __syncthreads();
    }
    if (threadIdx.x == 0) {
        part[(size_t)blockIdx.x * 3 + 0] = r0[0];
        part[(size_t)blockIdx.x * 3 + 1] = r1[0];
        part[(size_t)blockIdx.x * 3 + 2] = r2[0];
    }
}

__global__ __launch_bounds__(256) void loss_final_kernel(const double* __restrict__ part,
                                                         int nparts, float* __restrict__ out) {
    double sP = 0.0, sLogOff = 0.0, sLogDiag = 0.0;
    for (int b = threadIdx.x; b < nparts; b += 256) {
        sP += part[(size_t)b * 3 + 0];
        sLogOff += part[(size_t)b * 3 + 1];
        sLogDiag += part[(size_t)b * 3 + 2];
    }
    __shared__ double r0[256], r1[256], r2[256];
    r0[threadIdx.x] = sP; r1[threadIdx.x] = sLogOff; r2[threadIdx.x] = sLogDiag;
    __syncthreads();
    for (int off = 128; off; off >>= 1) {
        if (threadIdx.x < off) {
            r0[threadIdx.x] += r0[threadIdx.x + off];
            r1[threadIdx.x] += r1[threadIdx.x + off];
            r2[threadIdx.x] += r2[threadIdx.x + off];
        }
        __syncthreads();
    }
    if (threadIdx.x == 0) {
        const double nn = (double)NROWS * (double)NROWS;
        const double offcnt = nn - (double)NROWS;
        const double m = r0[0] / offcnt;                         // mean of off-diagonal Probs
        const double uni = (m * (offcnt * log(m) - r1[0])) / nn; // KL(Q||P), element mean
        const double liot = -r2[0] / (double)NROWS;              // 2N == NROWS
        out[0] = (float)(liot + (double)LAMBDA_P * uni);
    }
}

// ---------------- host orchestration ----------------
extern "C" void kernel_launch(void* const* d_in, const int* in_sizes, int n_in,
                              void* d_out, int out_size, void* d_ws, size_t ws_size,
                              hipStream_t stream) {
    (void)in_sizes; (void)n_in; (void)out_size; (void)ws_size;
    const float* z = (const float*)d_in[0];
    float* out = (float*)d_out;

    float* ws   = (float*)d_ws;
    float* zn   = ws;                                   // 8192*512
    float* P    = zn + (size_t)NROWS * KDIM;            // 8192*8192
    float* r    = P + (size_t)NROWS * NROWS;            // 8192
    float* ir   = r + NROWS;                            // 8192
    float* c    = ir + NROWS;                           // 8192
    float* ic   = c + NROWS;                            // 8192
    float* cpar = ic + NROWS;                           // 64*8192
    float* mpar = cpar + (size_t)64 * NROWS;            // 1024
    float* cmax = mpar + 1024;                          // 8 (pad for alignment)
    double* lpar = (double*)(cmax + 8);                 // 1024*3 doubles

    normalize_kernel<<<NROWS, 256, 0, stream>>>(z, zn);
    gemm_cos_kernel<<<dim3(16, 512), 256, 0, stream>>>(zn, P);

    min_partial_kernel<<<1024, 256, 0, stream>>>(P, mpar);
    min_final_kernel<<<1, 256, 0, stream>>>(mpar, cmax, 1024);
    exp_kernel<<<4096, 256, 0, stream>>>(P, cmax);

    // 10 Sinkhorn iterations; logical transpose tracked by iteration parity,
    // so the matrix is only ever scaled in place (no transposed writes).
    for (int it = 0; it < 10; ++it) {
        if ((it & 1) == 0) {  // logical == physical
            rowsum_kernel<<<NROWS, 256, 0, stream>>>(P, nullptr, r);
            recip_kernel<<<NROWS / 256, 256, 0, stream>>>(r, ir);
            colsum_partial_kernel<<<dim3(8, 64), 256, 0, stream>>>(P, ir, cpar);
            colsum_finish_kernel<<<8, 256, 0, stream>>>(cpar, c);
            recip_kernel<<<NROWS / 256, 256, 0, stream>>>(c, ic);
            scale_kernel<<<4096, 256, 0, stream>>>(P, ir, ic);   // P[i][j] /= r_i*c_j
        } else {              // logical == physical^T
            colsum_partial_kernel<<<dim3(8, 64), 256, 0, stream>>>(P, nullptr, cpar);
            colsum_finish_kernel<<<8, 256, 0, stream>>>(cpar, r);
            recip_kernel<<<NROWS / 256, 256, 0, stream>>>(r, ir);
            rowsum_kernel<<<NROWS, 256, 0, stream>>>(P, ir, c);  // c_j = sum_i P[j][i]/r_i
            recip_kernel<<<NROWS / 256, 256, 0, stream>>>(c, ic);
            scale_kernel<<<4096, 256, 0, stream>>>(P, ic, ir);   // P[a][b] /= c_a*r_b
        }
    }
    // after 10 transposes, logical orientation == physical again

    loss_partial_kernel<<<1024, 256, 0, stream>>>(P, lpar);
    loss_final_kernel<<<1, 256, 0, stream>>>(lpar, 1024, out);
}